// CachedParamMgr_82068235092460
// MI455X (gfx1250) — compile-verified
//
#include <hip/hip_runtime.h>
#include <stdint.h>

// ---------------------------------------------------------------------------
// CachedParamMgr cache-swap for MI455X (gfx1250).
// Pure bandwidth problem (~1.4 GB moved, 0 FLOPs -> ~60us @ 23.3 TB/s).
// CDNA5 paths used:
//   * global_load_async_to_lds_b128 / global_store_async_from_lds_b128
//     (ASYNCcnt-tracked DMA) for the dominant 512MB weight bulk copy,
//     pipelined with 2-parity LDS buffers + 4 chunks/thread so each wave
//     keeps 4 async loads in flight and overlaps stores with next loads.
//   * s_wait_asynccnt for completion
//   * b128 vector memory (float4) for all row scatter/gather
// top_k(masked_map, K) is implemented as histogram + exclusive scan +
// descending rank (values bounded in [-2, NUM_EMB)), which reproduces
// jax.lax.top_k exactly for unique values (true for these inputs).
// ---------------------------------------------------------------------------

#define DIM 128
#define ROW4 (DIM / 4)   // float4 per row
#define SCAN_B 1024
#define CPY_T 256        // threads per copy block
#define CPY_C 4          // chunks (float4) per thread per iteration

__device__ __forceinline__ void wait_asynccnt0() {
#if __has_builtin(__builtin_amdgcn_s_wait_asynccnt)
  __builtin_amdgcn_s_wait_asynccnt(0);
#else
  asm volatile("s_wait_asynccnt 0" ::: "memory");
#endif
}

__device__ __forceinline__ void async_g2l_b128(uint32_t lds_off, const void* g) {
  asm volatile("global_load_async_to_lds_b128 %0, %1, off"
               :: "v"(lds_off), "v"(g) : "memory");
}
__device__ __forceinline__ void async_l2g_b128(uint32_t lds_off, void* g) {
  asm volatile("global_store_async_from_lds_b128 %1, %0, off"
               :: "v"(lds_off), "v"(g) : "memory");
}

// --- phase 1: cpu_row_idxs = idx_map[ids]; mark backlist bitmap ------------
__global__ void k_gather_ids(const int* __restrict__ ids,
                             const int* __restrict__ idx_map,
                             int* __restrict__ cpu_rows,
                             unsigned* __restrict__ backlist, int K) {
  int j = blockIdx.x * blockDim.x + threadIdx.x;
  if (j < K) {
    int c = idx_map[ids[j]];
    cpu_rows[j] = c;
    atomicOr(&backlist[((unsigned)c) >> 5], 1u << (c & 31));
  }
}

// --- phase 2: masked eviction-score map + histogram ------------------------
__global__ void k_hist(const int* __restrict__ cached,
                       const unsigned* __restrict__ backlist,
                       int* __restrict__ masked, int* __restrict__ hist,
                       int rows, int num_emb) {
  int i = blockIdx.x * blockDim.x + threadIdx.x;
  if (i < rows) {
    int v = cached[i];
    if (v >= 0 && v < num_emb && ((backlist[v >> 5] >> (v & 31)) & 1u)) v = -2;
    masked[i] = v;
    int b = v;
    if (b < -2) b = -2;
    if (b > num_emb - 1) b = num_emb - 1;
    atomicAdd(&hist[b + 2], 1);
  }
}

// --- phase 3: exclusive scan of histogram (3-kernel two-level scan) --------
__global__ void k_scan1(const int* __restrict__ in, int* __restrict__ out,
                        int* __restrict__ partials, int n) {
  __shared__ int sm[SCAN_B];
  int t = threadIdx.x;
  int i = blockIdx.x * SCAN_B + t;
  int x = (i < n) ? in[i] : 0;
  sm[t] = x;
  __syncthreads();
  for (int off = 1; off < SCAN_B; off <<= 1) {
    int v = (t >= off) ? sm[t - off] : 0;
    __syncthreads();
    sm[t] += v;
    __syncthreads();
  }
  if (i < n) out[i] = sm[t] - x;  // exclusive within block
  if (t == SCAN_B - 1) partials[blockIdx.x] = sm[t];
}

__global__ void k_scan2(int* __restrict__ partials, int n) {
  __shared__ int sm[SCAN_B];
  int t = threadIdx.x;
  int running = 0;
  for (int base = 0; base < n; base += SCAN_B) {
    int i = base + t;
    int x = (i < n) ? partials[i] : 0;
    sm[t] = x;
    __syncthreads();
    for (int off = 1; off < SCAN_B; off <<= 1) {
      int v = (t >= off) ? sm[t - off] : 0;
      __syncthreads();
      sm[t] += v;
      __syncthreads();
    }
    int total = sm[SCAN_B - 1];
    if (i < n) partials[i] = running + sm[t] - x;
    running += total;
    __syncthreads();
  }
}

__global__ void k_scan3(int* __restrict__ out, const int* __restrict__ partials, int n) {
  int i = blockIdx.x * SCAN_B + threadIdx.x;
  if (i < n) out[i] += partials[blockIdx.x];
}

// --- phase 4: descending rank; rank r < K -> evict_slots[r] = slot ---------
__global__ void k_rank(const int* __restrict__ masked, const int* __restrict__ offs,
                       const int* __restrict__ hist, int* __restrict__ counters,
                       int* __restrict__ evict, int rows, int K, int num_emb) {
  int i = blockIdx.x * blockDim.x + threadIdx.x;
  if (i < rows) {
    int v = masked[i];
    if (v < -2) v = -2;
    if (v > num_emb - 1) v = num_emb - 1;
    int b = v + 2;
    // #elements strictly greater = rows - inclusive_count(<= b)
    int greater = rows - offs[b] - hist[b];
    int r = greater + atomicAdd(&counters[b], 1);
    if (r < K) evict[r] = i;
  }
}

// --- pipelined bulk copy via CDNA5 async global<->LDS DMA ------------------
// 2 parity buffers x 4 chunks/thread: issue 4 async loads (parity p), one
// s_wait_asynccnt 0 (also retires parity-p stores from 2 iters ago, already
// flushed by the previous wait), then 4 async stores (parity p) that overlap
// the NEXT iteration's loads into parity 1-p.
__global__ void k_copy16_async(const float4* __restrict__ src,
                               float4* __restrict__ dst, size_t n) {
  __shared__ float4 stage[2][CPY_C][CPY_T];
  const int t = threadIdx.x;
  const size_t per = (size_t)blockDim.x;                 // 256
  const size_t gstride = (size_t)gridDim.x * per * CPY_C;
  int p = 0;
  for (size_t base = (size_t)blockIdx.x * per * CPY_C; base < n; base += gstride) {
#pragma unroll
    for (int c = 0; c < CPY_C; ++c) {
      size_t idx = base + (size_t)c * per + t;
      if (idx < n)
        async_g2l_b128((uint32_t)(uintptr_t)&stage[p][c][t], src + idx);
    }
    wait_asynccnt0();  // parity-p loads done; all prior stores done too
#pragma unroll
    for (int c = 0; c < CPY_C; ++c) {
      size_t idx = base + (size_t)c * per + t;
      if (idx < n)
        async_l2g_b128((uint32_t)(uintptr_t)&stage[p][c][t], dst + idx);
    }
    p ^= 1;  // stores drain while next iteration loads the other parity
  }
  wait_asynccnt0();  // drain trailing stores before wave exit
}

// --- int32 map -> float32 output copy --------------------------------------
__global__ void k_i2f(const int* __restrict__ src, float* __restrict__ dst, int n) {
  int i = blockIdx.x * blockDim.x + threadIdx.x;
  if (i < n) dst[i] = (float)src[i];
}

// --- phase 5: write evicted rows back into weight_out; inverted[info]=-1 ---
__global__ void k_evict(const int* __restrict__ evict, const int* __restrict__ cached_in,
                        const float4* __restrict__ cache_in, float4* __restrict__ weight_out,
                        float* __restrict__ inv_out, int K, int num_emb) {
  int lane = threadIdx.x & 31;
  int r = (int)((blockIdx.x * blockDim.x + threadIdx.x) >> 5);  // one wave32 per row
  if (r >= K) return;
  int slot = evict[r];
  int info = cached_in[slot];
  if (info < 0 || info >= num_emb) return;
  weight_out[(size_t)info * ROW4 + lane] = cache_in[(size_t)slot * ROW4 + lane];
  if (lane == 0) inv_out[info] = -1.0f;
}

// --- phase 6: fetch rows into freed slots + emit output + update maps ------
__global__ void k_fetch(const int* __restrict__ evict, const int* __restrict__ cpu_rows,
                        const float4* __restrict__ weight_out_c,
                        float4* __restrict__ cache_out, float4* __restrict__ out_rows,
                        float* __restrict__ cachedmap_out, float* __restrict__ inv_out,
                        int K) {
  int lane = threadIdx.x & 31;
  int r = (int)((blockIdx.x * blockDim.x + threadIdx.x) >> 5);  // one wave32 per row
  if (r >= K) return;
  int slot = evict[r];
  int c = cpu_rows[r];
  float4 val = weight_out_c[(size_t)c * ROW4 + lane];
  cache_out[(size_t)slot * ROW4 + lane] = val;
  out_rows[(size_t)r * ROW4 + lane] = val;
  if (lane == 0) {
    cachedmap_out[slot] = (float)c;
    inv_out[c] = (float)slot;
  }
}

extern "C" void kernel_launch(void* const* d_in, const int* in_sizes, int n_in,
                              void* d_out, int out_size, void* d_ws, size_t ws_size,
                              hipStream_t stream) {
  const float* weight_in = (const float*)d_in[0];
  const float* cache_in  = (const float*)d_in[1];
  const int*   ids       = (const int*)d_in[2];
  const int*   idx_map   = (const int*)d_in[3];
  const int*   cached_in = (const int*)d_in[4];
  const int*   inv_in    = (const int*)d_in[5];

  const int num_emb = in_sizes[0] / DIM;       // 1,000,000
  const int rows    = in_sizes[1] / DIM;       // 200,000 (CUDA_ROWS)
  const int K       = in_sizes[2];             // 65,536
  const int nbins   = num_emb + 2;
  const int npart   = (nbins + SCAN_B - 1) / SCAN_B;

  // d_out layout: tuple concatenated flat, all float32
  float* out_rows      = (float*)d_out;                          // [K*DIM]
  float* cache_out     = out_rows + (size_t)K * DIM;             // [rows*DIM]
  float* weight_out    = cache_out + (size_t)rows * DIM;         // [num_emb*DIM]
  float* cachedmap_out = weight_out + (size_t)num_emb * DIM;     // [rows]
  float* inv_out       = cachedmap_out + rows;                   // [num_emb]

  // workspace layout (~13.5 MB)
  char* w = (char*)d_ws;
  int* cpu_rows = (int*)w;       w += sizeof(int) * (size_t)K;
  unsigned* backlist = (unsigned*)w; w += sizeof(unsigned) * (((size_t)num_emb + 31) / 32);
  int* masked = (int*)w;         w += sizeof(int) * (size_t)rows;
  int* hist = (int*)w;           w += sizeof(int) * (size_t)nbins;
  int* offs = (int*)w;           w += sizeof(int) * (size_t)nbins;
  int* counters = (int*)w;       w += sizeof(int) * (size_t)nbins;
  int* partials = (int*)w;       w += sizeof(int) * (size_t)npart;
  int* evict = (int*)w;          w += sizeof(int) * (size_t)K;

  // zero the accumulated arrays (graph-capture-safe memset nodes)
  hipMemsetAsync(backlist, 0, sizeof(unsigned) * (((size_t)num_emb + 31) / 32), stream);
  hipMemsetAsync(hist, 0, sizeof(int) * (size_t)nbins, stream);
  hipMemsetAsync(counters, 0, sizeof(int) * (size_t)nbins, stream);

  const int T256 = 256;

  // 1) gather + backlist
  k_gather_ids<<<(K + T256 - 1) / T256, T256, 0, stream>>>(ids, idx_map, cpu_rows, backlist, K);
  // 2) masked map + histogram
  k_hist<<<(rows + T256 - 1) / T256, T256, 0, stream>>>(cached_in, backlist, masked, hist, rows, num_emb);
  // 3) exclusive scan of histogram
  k_scan1<<<npart, SCAN_B, 0, stream>>>(hist, offs, partials, nbins);
  k_scan2<<<1, SCAN_B, 0, stream>>>(partials, npart);
  k_scan3<<<npart, SCAN_B, 0, stream>>>(offs, partials, nbins);
  // 4) descending rank -> evict slot list (top_k indices)
  k_rank<<<(rows + T256 - 1) / T256, T256, 0, stream>>>(masked, offs, hist, counters, evict, rows, K, num_emb);

  // 5) bulk copies of the returned tensors (dominant traffic; async-DMA path)
  k_copy16_async<<<2048, CPY_T, 0, stream>>>((const float4*)weight_in, (float4*)weight_out,
                                             (size_t)num_emb * ROW4);
  k_copy16_async<<<1024, CPY_T, 0, stream>>>((const float4*)cache_in, (float4*)cache_out,
                                             (size_t)rows * ROW4);
  k_i2f<<<(rows + T256 - 1) / T256, T256, 0, stream>>>(cached_in, cachedmap_out, rows);
  k_i2f<<<(num_emb + T256 - 1) / T256, T256, 0, stream>>>(inv_in, inv_out, num_emb);

  // 6) evict: cache rows -> weight_out, inverted[evict_info] = -1
  k_evict<<<(K * 32 + T256 - 1) / T256, T256, 0, stream>>>(evict, cached_in,
      (const float4*)cache_in, (float4*)weight_out, inv_out, K, num_emb);
  // 7) fetch: weight_out rows -> cache_out + out; update both maps
  k_fetch<<<(K * 32 + T256 - 1) / T256, T256, 0, stream>>>(evict, cpu_rows,
      (const float4*)weight_out, (float4*)cache_out, (float4*)out_rows,
      cachedmap_out, inv_out, K);

  (void)n_in; (void)out_size; (void)ws_size;
}